// DRR_12240656794305
// MI455X (gfx1250) — compile-verified
//
#include <hip/hip_runtime.h>
#include <hip/hip_bf16.h>
#include <math.h>

typedef __attribute__((ext_vector_type(2))) float v2f;
typedef __attribute__((ext_vector_type(8))) float v8f;

namespace {
constexpr int   kVol = 256;
constexpr int   kH = 160;
constexpr int   kW = 160;
constexpr float kSDD = 1020.0f;
constexpr float kDelX = 2.5f;
constexpr float kDelY = 2.5f;
constexpr float kEps = 1e-8f;
}

// One wave (32 lanes) per block; each lane integrates one ray via incremental
// Siddon DDA.  Ray setup (pose + affine transforms) is done as a batched 4x4
// matmul on the WMMA unit: D(16x16) = A(16x4) x B(4x16).
__global__ __launch_bounds__(32) void drr_siddon_wmma_kernel(
    const float* __restrict__ vol,    // [256^3]
    const float* __restrict__ pose,   // [1,4,4] row-major
    const float* __restrict__ ainv,   // [4,4] row-major (affine_inv)
    float* __restrict__ out)          // [25600]
{
    const int lane = threadIdx.x;           // 0..31
    const int base = blockIdx.x * 32;       // first ray of this wave

    // ---------------- WMMA A-matrix (16x4): transform rows -----------------
    // rows 0..2 : Mc = (affine_inv @ pose) rows (voxel-space transform)
    // rows 3..5 : pose rotation rows with 0 translation (world-space delta)
    // rows 8..13: duplicate of rows 0..5 (so lanes 16..31 read D directly)
    // A layout: lane l holds row (l&15); VGPR0 = K (2*(l>>4)), VGPR1 = K+1.
    const int rr = lane & 15;
    const int re = rr & 7;                  // row within the duplicated group
    const int kp = lane >> 4;
    const int k0 = kp * 2, k1 = kp * 2 + 1;

    float ax = 0.0f, ay = 0.0f;
    if (re < 3) {
        // Mc[re][k] = sum_j ainv[re][j] * pose[j][k]   (pose row3 = [0,0,0,1])
        float m0 = 0.0f, m1 = 0.0f;
        #pragma unroll
        for (int j = 0; j < 4; ++j) {
            const float av = ainv[re * 4 + j];
            m0 = fmaf(av, pose[j * 4 + k0], m0);
            m1 = fmaf(av, pose[j * 4 + k1], m1);
        }
        ax = m0; ay = m1;
    } else if (re < 6) {
        const int i = re - 3;               // rotation-only rows, no translation
        ax = (k0 < 3) ? pose[i * 4 + k0] : 0.0f;
        ay = (k1 < 3) ? pose[i * 4 + k1] : 0.0f;
    }
    v2f A; A.x = ax; A.y = ay;

    // Source in voxel coords: s[i] = Mc[i][3]
    float s0 = 0.0f, s1 = 0.0f, s2 = 0.0f;
    #pragma unroll
    for (int j = 0; j < 4; ++j) {
        const float pj = pose[j * 4 + 3];   // pose[3][3] = 1 -> homogeneous
        s0 = fmaf(ainv[0 * 4 + j], pj, s0);
        s1 = fmaf(ainv[1 * 4 + j], pj, s1);
        s2 = fmaf(ainv[2 * 4 + j], pj, s2);
    }

    // ---------------- WMMA B-matrices (4x16): detector points --------------
    // B layout: lane l holds column (l&15); VGPR0 = K (2*(l>>4)), VGPR1 = K+1.
    // K components of homogeneous point: {tx, ty, SDD, 1}.
    const int  nl = lane & 15;
    const bool lo = (lane < 16);

    const int n1 = base + nl;
    const int n2 = base + 16 + nl;
    const float tx1 = ((n1 % kW) - (kW - 1) * 0.5f) * kDelX;
    const float ty1 = ((n1 / kW) - (kH - 1) * 0.5f) * kDelY;
    const float tx2 = ((n2 % kW) - (kW - 1) * 0.5f) * kDelX;
    const float ty2 = ((n2 / kW) - (kH - 1) * 0.5f) * kDelY;

    v2f B1; B1.x = lo ? tx1 : kSDD; B1.y = lo ? ty1 : 1.0f;
    v2f B2; B2.x = lo ? tx2 : kSDD; B2.y = lo ? ty2 : 1.0f;

    v8f c0 = {};
    // D = A x B + 0  (16x16 f32 accumulators, 8 VGPRs)
    v8f D1 = __builtin_amdgcn_wmma_f32_16x16x4_f32(
        false, A, false, B1, (short)0, c0, false, false);
    v8f D2 = __builtin_amdgcn_wmma_f32_16x16x4_f32(
        false, A, false, B2, (short)0, c0, false, false);

    // D layout: VGPR d, lanes 0..15 -> row d, lanes 16..31 -> row d+8.
    // Rows 8..13 duplicate 0..5, so lane l>=16 reads its own ray from D2.
    const float tv0 = lo ? D1[0] : D2[0];   // target, voxel coords
    const float tv1 = lo ? D1[1] : D2[1];
    const float tv2 = lo ? D1[2] : D2[2];
    const float dw0 = lo ? D1[3] : D2[3];   // target - source, world coords
    const float dw1 = lo ? D1[4] : D2[4];
    const float dw2 = lo ? D1[5] : D2[5];

    const float raylen = sqrtf(fmaf(dw0, dw0, fmaf(dw1, dw1, dw2 * dw2)));

    // ---------------- exact Siddon via incremental DDA ---------------------
    const float sd0 = tv0 - s0, sd1 = tv1 - s1, sd2 = tv2 - s2;
    const float sds0 = (fabsf(sd0) < kEps) ? kEps : sd0;
    const float sds1 = (fabsf(sd1) < kEps) ? kEps : sd1;
    const float sds2 = (fabsf(sd2) < kEps) ? kEps : sd2;

    const float dims = (float)kVol;
    float amin = 0.0f, amax = 1.0f;
    {
        const float a00 = (0.0f - s0) / sds0, a01 = (dims - s0) / sds0;
        const float a10 = (0.0f - s1) / sds1, a11 = (dims - s1) / sds1;
        const float a20 = (0.0f - s2) / sds2, a21 = (dims - s2) / sds2;
        amin = fmaxf(amin, fminf(a00, a01));
        amin = fmaxf(amin, fminf(a10, a11));
        amin = fmaxf(amin, fminf(a20, a21));
        amax = fminf(amax, fmaxf(a00, a01));
        amax = fminf(amax, fmaxf(a10, a11));
        amax = fminf(amax, fmaxf(a20, a21));
        amax = fmaxf(amax, amin);
    }

    const float inv0 = 1.0f / sds0, inv1 = 1.0f / sds1, inv2 = 1.0f / sds2;
    const float dt0 = fabsf(inv0), dt1 = fabsf(inv1), dt2 = fabsf(inv2);

    // first plane crossing strictly after amin, per axis
    float p0 = fmaf(amin, sds0, s0);
    float p1 = fmaf(amin, sds1, s1);
    float p2 = fmaf(amin, sds2, s2);
    float tn0 = (sds0 > 0.0f) ? (floorf(p0) + 1.0f - s0) * inv0
                              : (ceilf(p0) - 1.0f - s0) * inv0;
    float tn1 = (sds1 > 0.0f) ? (floorf(p1) + 1.0f - s1) * inv1
                              : (ceilf(p1) - 1.0f - s1) * inv1;
    float tn2 = (sds2 > 0.0f) ? (floorf(p2) + 1.0f - s2) * inv2
                              : (ceilf(p2) - 1.0f - s2) * inv2;
    tn0 = fmaxf(tn0, amin);
    tn1 = fmaxf(tn1, amin);
    tn2 = fmaxf(tn2, amin);

    // warm the entry voxel (global_prefetch_b8); volume is L2-resident (64MB)
    {
        int j0 = min(max((int)floorf(p0), 0), kVol - 1);
        int j1 = min(max((int)floorf(p1), 0), kVol - 1);
        int j2 = min(max((int)floorf(p2), 0), kVol - 1);
        __builtin_prefetch(vol + ((((j0 << 8) + j1) << 8) + j2), 0, 3);
    }

    float acc = 0.0f;
    float t = amin;
    int it = 0;
    while (t < amax && it < 2048) {
        ++it;
        const float tn = fminf(fminf(tn0, tn1), fminf(tn2, amax));
        const float tm = 0.5f * (t + tn);
        // midpoint voxel index (same formula as reference: floor(s + amid*sd))
        int i0 = (int)floorf(fmaf(tm, sd0, s0));
        int i1 = (int)floorf(fmaf(tm, sd1, s1));
        int i2 = (int)floorf(fmaf(tm, sd2, s2));
        i0 = min(max(i0, 0), kVol - 1);
        i1 = min(max(i1, 0), kVol - 1);
        i2 = min(max(i2, 0), kVol - 1);
        acc = fmaf(vol[(((i0 << 8) + i1) << 8) + i2], tn - t, acc);
        t = tn;
        if (tn0 <= tn) tn0 += dt0;
        if (tn1 <= tn) tn1 += dt1;
        if (tn2 <= tn) tn2 += dt2;
    }

    out[base + lane] = acc * raylen;
}

extern "C" void kernel_launch(void* const* d_in, const int* in_sizes, int n_in,
                              void* d_out, int out_size, void* d_ws, size_t ws_size,
                              hipStream_t stream) {
    const float* vol  = (const float*)d_in[0];   // density [256,256,256] f32
    const float* pose = (const float*)d_in[1];   // [1,4,4] f32
    const float* ainv = (const float*)d_in[2];   // [4,4]   f32
    float* out = (float*)d_out;                  // [1,1,160,160] f32

    const int rays = kH * kW;                    // 25600
    dim3 grid(rays / 32);                        // 800 single-wave workgroups
    dim3 block(32);
    hipLaunchKernelGGL(drr_siddon_wmma_kernel, grid, block, 0, stream,
                       vol, pose, ainv, out);
}